// STSMPN_16612933501120
// MI455X (gfx1250) — compile-verified
//
#include <hip/hip_runtime.h>

// Problem dims (fixed by the reference)
#define BB 16
#define SS 12
#define NN 207
#define DD 128
#define CC 8
#define LL 2
#define PP 6

typedef float v2f __attribute__((ext_vector_type(2)));
typedef float v4f __attribute__((ext_vector_type(4)));
typedef float v8f __attribute__((ext_vector_type(8)));

// ---------------------------------------------------------------------------
// Gather: feats[b,p,n2,d] = inputs[b, ckps[p][n2>=N], n2%N, d]
// ---------------------------------------------------------------------------
__global__ __launch_bounds__(256) void gather_feats_kernel(
    const float* __restrict__ inputs, const int* __restrict__ ckps,
    float* __restrict__ feats, int total4)
{
    int tid = blockIdx.x * blockDim.x + threadIdx.x;
    if (tid >= total4) return;
    int d4   = tid & 31;              // DD/4 = 32 float4 per row
    int rest = tid >> 5;
    int n2   = rest % (2 * NN);
    rest    /= (2 * NN);
    int p    = rest % PP;
    int b    = rest / PP;
    int half = (n2 >= NN) ? 1 : 0;
    int n    = n2 - half * NN;
    int s    = ckps[p * 2 + half];
    const v4f* src = (const v4f*)(inputs + (((size_t)(b * SS + s) * NN + n) * DD)) + d4;
    v4f*       dst = (v4f*)(feats + (((size_t)(b * PP + p) * (2 * NN) + n2) * DD)) + d4;
    *dst = *src;
}

// ---------------------------------------------------------------------------
// Batched fp32 GEMM via V_WMMA_F32_16X16X4_F32, register-blocked.
// One wave computes a 64x64 block of C = A(MxK) * B(KxNc): 4x4 tiles of
// 16x16, 16 v8f accumulators. Per K-step of 4: 4 A frags + 4 B frags feed
// 16 WMMAs (4x data reuse in both operands).
// Requires Nc % 64 == 0 (always 128 here). M edges: A row pointers are
// clamped (garbage rows only affect unstored D rows); stores are guarded.
// biasMode: 0 = none, 1 = per-column bias[col], 2 = per-row bias[row]
// ---------------------------------------------------------------------------
__global__ __launch_bounds__(32) void gemm_wmma_f32(
    const float* __restrict__ Abase, const float* __restrict__ Bbase,
    float* __restrict__ Cbase,
    int M, int K, int Nc,
    int lda, int ldb, int ldc,
    long long sA, long long sB, long long sC,
    const float* __restrict__ bias, int biasMode, int doRelu)
{
    const int lane  = threadIdx.x;        // 0..31
    const int half  = lane >> 4;          // 0 or 1
    const int l16   = lane & 15;
    const int tileN = blockIdx.x * 64;
    const int tileM = blockIdx.y * 64;
    const int batch = blockIdx.z;

    const float* A  = Abase + (size_t)batch * sA;
    const float* Bm = Bbase + (size_t)batch * sB;
    float*       Cm = Cbase + (size_t)batch * sC;

    const int koff = half * 2;            // this lane-half covers K = k0+koff, k0+koff+1

    // Hoisted, clamped per-subtile A row pointers.
    const float* Aptr[4];
#pragma unroll
    for (int i = 0; i < 4; ++i) {
        const int arow = tileM + 16 * i + l16;
        Aptr[i] = A + (size_t)(arow < M ? arow : (M - 1)) * lda + koff;
    }
    // B base pointer for this lane: row koff, column tileN + l16.
    const float* Bptr = Bm + (size_t)koff * ldb + (tileN + l16);

    v8f acc[4][4] = {};

    const int Kmain = K & ~3;
    for (int k0 = 0; k0 < Kmain; k0 += 4) {
        v2f a[4], b[4];
#pragma unroll
        for (int i = 0; i < 4; ++i) {
            a[i].x = Aptr[i][k0];
            a[i].y = Aptr[i][k0 + 1];
        }
        const float* bp = Bptr + (size_t)k0 * ldb;
#pragma unroll
        for (int j = 0; j < 4; ++j) {
            b[j].x = bp[16 * j];
            b[j].y = bp[(size_t)ldb + 16 * j];
        }
#pragma unroll
        for (int i = 0; i < 4; ++i)
#pragma unroll
            for (int j = 0; j < 4; ++j)
                acc[i][j] = __builtin_amdgcn_wmma_f32_16x16x4_f32(
                    false, a[i], false, b[j], (short)0, acc[i][j], false, false);
    }
    if (Kmain < K) {                       // zero-padded K tail (remainder 1..3)
        const int  ka  = Kmain + koff;
        const bool ok0 = (ka < K), ok1 = (ka + 1 < K);
        v2f a[4], b[4];
#pragma unroll
        for (int i = 0; i < 4; ++i) {
            a[i].x = ok0 ? Aptr[i][Kmain]     : 0.f;
            a[i].y = ok1 ? Aptr[i][Kmain + 1] : 0.f;
        }
        const float* bp = Bptr + (size_t)Kmain * ldb;
#pragma unroll
        for (int j = 0; j < 4; ++j) {
            b[j].x = ok0 ? bp[16 * j]               : 0.f;
            b[j].y = ok1 ? bp[(size_t)ldb + 16 * j] : 0.f;
        }
#pragma unroll
        for (int i = 0; i < 4; ++i)
#pragma unroll
            for (int j = 0; j < 4; ++j)
                acc[i][j] = __builtin_amdgcn_wmma_f32_16x16x4_f32(
                    false, a[i], false, b[j], (short)0, acc[i][j], false, false);
    }

    // Epilogue: bias + relu + guarded store.
#pragma unroll
    for (int i = 0; i < 4; ++i) {
#pragma unroll
        for (int r = 0; r < 8; ++r) {
            const int row = tileM + 16 * i + r + 8 * half;
            if (row < M) {
                float rowb = (biasMode == 2) ? bias[row] : 0.f;
#pragma unroll
                for (int j = 0; j < 4; ++j) {
                    const int col = tileN + 16 * j + l16;
                    float v = acc[i][j][r];
                    if (biasMode == 1) v += bias[col];
                    v += rowb;
                    if (doRelu) v = v > 0.f ? v : 0.f;
                    Cm[(size_t)row * ldc + col] = v;
                }
            }
        }
    }
}

// ---------------------------------------------------------------------------
// Host orchestration
// ---------------------------------------------------------------------------
static inline void launch_gemm(hipStream_t stream,
                               const float* A, const float* Bm, float* C,
                               int M, int K, int Nc,
                               int lda, int ldb, int ldc,
                               long long sA, long long sB, long long sC, int batch,
                               const float* bias, int biasMode, int relu)
{
    dim3 grid((Nc + 63) / 64, (M + 63) / 64, batch);
    gemm_wmma_f32<<<grid, dim3(32), 0, stream>>>(A, Bm, C, M, K, Nc, lda, ldb, ldc,
                                                 sA, sB, sC, bias, biasMode, relu);
}

extern "C" void kernel_launch(void* const* d_in, const int* in_sizes, int n_in,
                              void* d_out, int out_size, void* d_ws, size_t ws_size,
                              hipStream_t stream)
{
    const float* inputs = (const float*)d_in[0];
    const float* A_back = (const float*)d_in[1];
    const float* A_fwd  = (const float*)d_in[2];
    const float* gcn_W  = (const float*)d_in[3];   // (C,L,D,D)
    const float* gcn_b  = (const float*)d_in[4];   // (C,L,D)
    const float* conv_W = (const float*)d_in[5];   // (N,2N)
    const float* conv_b = (const float*)d_in[6];   // (N,)
    const float* lin_W  = (const float*)d_in[7];   // (C*D,D)
    const float* lin_b  = (const float*)d_in[8];   // (D,)
    const int*   ckps   = (const int*)d_in[9];     // (P,2)
    float* out = (float*)d_out;

    const int    BP       = BB * PP;                       // 96
    const size_t feats_sz = (size_t)BP * 2 * NN * DD;      // 5,087,232 floats
    const size_t res_sz   = (size_t)BP * NN * CC * DD;     // 20,348,928 floats

    float* feats = (float*)d_ws;          // (B,P,2N,D)
    float* t0    = feats + feats_sz;      // shared layer-1 diffusion output
    float* bufA  = t0 + feats_sz;         // per-channel h
    float* bufB  = bufA + feats_sz;       // per-channel diffusion
    float* res   = bufB + feats_sz;       // (B,P,N,C,D)

    // 1) gather checkpoint features
    {
        int total4 = BP * 2 * NN * (DD / 4);
        gather_feats_kernel<<<(total4 + 255) / 256, 256, 0, stream>>>(
            inputs, ckps, feats, total4);
    }

    // 2) layer-1 diffusion (channel-independent: compute once)
    launch_gemm(stream, A_back, feats,            t0,
                NN, NN, DD, NN, DD, DD,
                0, (long long)2 * NN * DD, (long long)2 * NN * DD, BP,
                nullptr, 0, 0);
    launch_gemm(stream, A_fwd,  feats + (size_t)NN * DD, t0 + (size_t)NN * DD,
                NN, NN, DD, NN, DD, DD,
                0, (long long)2 * NN * DD, (long long)2 * NN * DD, BP,
                nullptr, 0, 0);

    // 3) per-channel pipeline
    for (int c = 0; c < CC; ++c) {
        const float* W0 = gcn_W + ((size_t)(c * LL + 0)) * DD * DD;
        const float* W1 = gcn_W + ((size_t)(c * LL + 1)) * DD * DD;
        const float* b0 = gcn_b + ((size_t)(c * LL + 0)) * DD;
        const float* b1 = gcn_b + ((size_t)(c * LL + 1)) * DD;

        // h1_c = relu(t0 @ W0 + b0)        (batched over (b,p), M=2N, K=D)
        launch_gemm(stream, t0, W0, bufA,
                    2 * NN, DD, DD, DD, DD, DD,
                    (long long)2 * NN * DD, 0, (long long)2 * NN * DD, BP,
                    b0, 1, 1);

        // t1_c = [A_back @ h1_top ; A_fwd @ h1_bot]
        launch_gemm(stream, A_back, bufA, bufB,
                    NN, NN, DD, NN, DD, DD,
                    0, (long long)2 * NN * DD, (long long)2 * NN * DD, BP,
                    nullptr, 0, 0);
        launch_gemm(stream, A_fwd, bufA + (size_t)NN * DD, bufB + (size_t)NN * DD,
                    NN, NN, DD, NN, DD, DD,
                    0, (long long)2 * NN * DD, (long long)2 * NN * DD, BP,
                    nullptr, 0, 0);

        // h2_c = relu(t1_c @ W1 + b1)   (overwrite bufA)
        launch_gemm(stream, bufB, W1, bufA,
                    2 * NN, DD, DD, DD, DD, DD,
                    (long long)2 * NN * DD, 0, (long long)2 * NN * DD, BP,
                    b1, 1, 1);

        // conv block: res[b,p,o, c*D + e] = sum_n conv_W[o,n] h2_c[b,p,n,e] + conv_b[o]
        launch_gemm(stream, conv_W, bufA, res + (size_t)c * DD,
                    NN, 2 * NN, DD, 2 * NN, DD, CC * DD,
                    0, (long long)2 * NN * DD, (long long)NN * CC * DD, BP,
                    conv_b, 2, 0);
    }

    // 4) final linear: out = res(B*P*N, C*D) @ lin_W + lin_b
    launch_gemm(stream, res, lin_W, out,
                BP * NN, CC * DD, DD, CC * DD, DD, DD,
                0, 0, 0, 1,
                lin_b, 1, 0);

    (void)in_sizes; (void)n_in; (void)out_size; (void)ws_size; (void)res_sz;
}